// HumanComposer3D_86500641341770
// MI455X (gfx1250) — compile-verified
//
#include <hip/hip_runtime.h>
#include <cstdint>

// ---------------------------------------------------------------------------
// HumanComposer3D for MI455X (gfx1250).
// Memory-bound streaming kernel: ~345 B/pixel of HBM traffic, ~18us floor at
// 23.3 TB/s. No matrix structure -> no WMMA; instead we exercise the CDNA5
// async global->LDS path (ASYNCcnt) to stage the 160B/pixel texel records,
// and non-temporal loads/stores to keep the one-pass stream out of L2.
// ---------------------------------------------------------------------------

#define TPB      256
#define KL       8      // layers / classes
#define TEXF     40     // floats per pixel in texels (8 layers x 5)
#define TEXPAD   44     // padded LDS stride in floats (16B aligned, fewer bank conflicts)

typedef float f4  __attribute__((ext_vector_type(4)));
typedef int   v4i __attribute__((ext_vector_type(4)));

#if defined(__has_builtin)
#  if __has_builtin(__builtin_amdgcn_global_load_async_to_lds_b128)
#    define HAVE_ASYNC_B128 1
#  endif
#  if __has_builtin(__builtin_amdgcn_s_wait_asynccnt)
#    define HAVE_WAIT_ASYNC 1
#  endif
#endif

__global__ __launch_bounds__(TPB)
void HumanComposer3D_86500641341770_kernel(const float* __restrict__ texels,
                                           const float* __restrict__ zbuf,
                                           const float* __restrict__ bkg,
                                           float* __restrict__ out,
                                           int P)
{
    __shared__ float smem[TPB * TEXPAD];

    const int tid = threadIdx.x;
    const int pix = blockIdx.x * TPB + tid;
    if (pix >= P) return;

    const float* gsrc = texels + (size_t)pix * TEXF;
    float*       lbuf = smem + tid * TEXPAD;

    // ---- Stage this lane's 40 texel floats into LDS: 10x async b128 --------
#if HAVE_ASYNC_B128
    #pragma unroll
    for (int j = 0; j < 10; ++j) {
        __builtin_amdgcn_global_load_async_to_lds_b128(
            (__attribute__((address_space(1))) v4i*)(gsrc + j * 4),
            (__attribute__((address_space(3))) v4i*)(lbuf + j * 4),
            /*offset=*/0, /*cpol=*/0);
    }
#else
    #pragma unroll
    for (int j = 0; j < 10; ++j) {
        *(f4*)(lbuf + j * 4) = *(const f4*)(gsrc + j * 4);
    }
#endif

    // ---- zbuf: 8 floats, non-temporal direct loads -------------------------
    const f4* zv = (const f4*)(zbuf + (size_t)pix * KL);
    f4 zlo = __builtin_nontemporal_load(zv + 0);
    f4 zhi = __builtin_nontemporal_load(zv + 1);
    float z[KL];
    #pragma unroll
    for (int k = 0; k < 4; ++k) { z[k] = zlo[k]; z[k + 4] = zhi[k]; }

    const float br = bkg[0], bg = bkg[1], bb = bkg[2];

    // ---- Wait for this wave's async loads (no cross-lane LDS sharing) ------
#if HAVE_ASYNC_B128
#  if HAVE_WAIT_ASYNC
    __builtin_amdgcn_s_wait_asynccnt(0);
#  else
    asm volatile("s_wait_asynccnt 0" ::: "memory");
#  endif
#endif

    // ---- Pull the 40 staged floats back as 10x ds_load_b128 ----------------
    f4 t4[10];
    #pragma unroll
    for (int j = 0; j < 10; ++j) t4[j] = *(const f4*)(lbuf + j * 4);
    const float* tex = (const float*)t4;   // fully unrolled constant indexing

    float cr[KL], cg[KL], cb[KL], ca[KL];
    int   li[KL];
    #pragma unroll
    for (int k = 0; k < KL; ++k) {
        cr[k] = tex[k * 5 + 0];
        cg[k] = tex[k * 5 + 1];
        cb[k] = tex[k * 5 + 2];
        ca[k] = tex[k * 5 + 3];
        li[k] = (z[k] >= 0.0f) ? __float2int_rn(tex[k * 5 + 4]) : -1;
    }

    // ---- Back-to-front compositing scan (k = 7 .. 0) -----------------------
    float r = br, g = bg, b = bb, a = 0.0f, d = 100.0f, l = (float)KL;
    #pragma unroll
    for (int k = KL - 1; k >= 0; --k) {
        const float la = ca[k], ld = z[k];
        const float om = 1.0f - la;
        r = fmaf(cr[k], la, r * om);
        g = fmaf(cg[k], la, g * om);
        b = fmaf(cb[k], la, b * om);
        a = fmaxf(la, a);
        d = (ld > 0.0f) ? fmaf(ld, la, d * om) : d;
        l = (ld >= 0.0f && la > 0.5f) ? tex[k * 5 + 4] : l;
    }

    const size_t Pz = (size_t)P;

    // composite_image: [P,4]
    f4 img = { r, g, b, a };
    __builtin_nontemporal_store(img, (f4*)out + pix);
    // composite_depth: [P]
    __builtin_nontemporal_store(d, out + 4 * Pz + pix);
    // composite_label: [P]  (exact-integer float; K -> -1)
    float lab = (l > (float)KL - 0.5f) ? -1.0f : l;
    __builtin_nontemporal_store(lab, out + 5 * Pz + pix);

    // ---- human_images: per class n, first layer k with label==n ------------
    f4* hout = (f4*)(out + 6 * Pz) + (size_t)pix * KL;
    #pragma unroll
    for (int n = 0; n < KL; ++n) {
        float hr = 0.0f, hg = 0.0f, hb = 0.0f, ha = 0.0f;
        // descending overwrite => keeps the SMALLEST matching k (branchless)
        #pragma unroll
        for (int k = KL - 1; k >= 0; --k) {
            const bool m = (li[k] == n);
            hr = m ? cr[k] : hr;
            hg = m ? cg[k] : hg;
            hb = m ? cb[k] : hb;
            ha = m ? ca[k] : ha;
        }
        const float om = 1.0f - ha;
        f4 o = { fmaf(hr, ha, br * om),
                 fmaf(hg, ha, bg * om),
                 fmaf(hb, ha, bb * om),
                 ha };
        __builtin_nontemporal_store(o, hout + n);
    }
}

extern "C" void kernel_launch(void* const* d_in, const int* in_sizes, int n_in,
                              void* d_out, int out_size, void* d_ws, size_t ws_size,
                              hipStream_t stream) {
    (void)n_in; (void)out_size; (void)d_ws; (void)ws_size;
    const float* texels = (const float*)d_in[0];   // [B,H,W,K,5] f32
    const float* zbuf   = (const float*)d_in[1];   // [B,H,W,K]   f32
    const float* bkg    = (const float*)d_in[2];   // [3]         f32
    float* out = (float*)d_out;                    // 38*P floats

    const int P = in_sizes[0] / TEXF;              // B*H*W = 1,179,648
    const int blocks = (P + TPB - 1) / TPB;        // 4608
    hipLaunchKernelGGL(HumanComposer3D_86500641341770_kernel,
                       dim3(blocks), dim3(TPB), 0, stream,
                       texels, zbuf, bkg, out, P);
}